// MLP_15247133901514
// MI455X (gfx1250) — compile-verified
//
#include <hip/hip_runtime.h>
#include <math.h>

typedef float v2f __attribute__((ext_vector_type(2)));
typedef float v8f __attribute__((ext_vector_type(8)));
typedef unsigned int u32x4 __attribute__((ext_vector_type(4)));
typedef int i32x4 __attribute__((ext_vector_type(4)));
typedef int i32x8 __attribute__((ext_vector_type(8)));

#define NSAMP 512
#define KDIM  1024
#define HIDD  512

// ---------------------------------------------------------------------------
// Kernel 1: C[i,j] = sum_k X[i,k] * W[j,k] (+ bias[j] for z==0)
// One wave computes a 16x64 strip of C using V_WMMA_F32_16X16X4_F32.
// ---------------------------------------------------------------------------
__global__ __launch_bounds__(32) void gemm_wmma_f32(
    const float* __restrict__ X,
    const float* __restrict__ Wh,
    const float* __restrict__ Wm,
    const float* __restrict__ bh,
    float* __restrict__ Hout,
    float* __restrict__ Mout)
{
    const int lane = threadIdx.x & 31;
    const int half = lane >> 4;      // 0: K lo pair, 1: K hi pair
    const int l16  = lane & 15;
    const int i0 = blockIdx.y * 16;
    const int j0 = blockIdx.x * 64;
    const bool isH = (blockIdx.z == 0);

    const float* W   = isH ? Wh : Wm;
    float*       Out = isH ? Hout : Mout;

    const float* arow  = X + (size_t)(i0 + l16) * KDIM + 2 * half;
    const float* brow0 = W + (size_t)(j0 +  0 + l16) * KDIM + 2 * half;
    const float* brow1 = W + (size_t)(j0 + 16 + l16) * KDIM + 2 * half;
    const float* brow2 = W + (size_t)(j0 + 32 + l16) * KDIM + 2 * half;
    const float* brow3 = W + (size_t)(j0 + 48 + l16) * KDIM + 2 * half;

    v8f acc0 = {}, acc1 = {}, acc2 = {}, acc3 = {};

#pragma unroll 4
    for (int k0 = 0; k0 < KDIM; k0 += 4) {
        v2f a  = *(const v2f*)(arow  + k0);
        v2f b0 = *(const v2f*)(brow0 + k0);
        v2f b1 = *(const v2f*)(brow1 + k0);
        v2f b2 = *(const v2f*)(brow2 + k0);
        v2f b3 = *(const v2f*)(brow3 + k0);
        acc0 = __builtin_amdgcn_wmma_f32_16x16x4_f32(false, a, false, b0, (short)0, acc0, false, false);
        acc1 = __builtin_amdgcn_wmma_f32_16x16x4_f32(false, a, false, b1, (short)0, acc1, false, false);
        acc2 = __builtin_amdgcn_wmma_f32_16x16x4_f32(false, a, false, b2, (short)0, acc2, false, false);
        acc3 = __builtin_amdgcn_wmma_f32_16x16x4_f32(false, a, false, b3, (short)0, acc3, false, false);
    }

    if (isH) {  // bias depends on column N = l16 only -> splat across all 8 rows
        acc0 += bh[j0 +  0 + l16];
        acc1 += bh[j0 + 16 + l16];
        acc2 += bh[j0 + 32 + l16];
        acc3 += bh[j0 + 48 + l16];
    }

    // C/D layout: lanes 0-15 -> M = vgpr, N = lane; lanes 16-31 -> M = 8+vgpr, N = lane-16
    const int rbase = i0 + 8 * half;
#pragma unroll
    for (int r = 0; r < 8; ++r) {
        const size_t row = (size_t)(rbase + r) * HIDD;
        Out[row + j0 +  0 + l16] = acc0[r];
        Out[row + j0 + 16 + l16] = acc1[r];
        Out[row + j0 + 32 + l16] = acc2[r];
        Out[row + j0 + 48 + l16] = acc3[r];
    }
}

// ---------------------------------------------------------------------------
// TDM: issue one tensor_load_to_lds for a 2D tile (f32 elements).
//   lds_addr     : LDS byte address of destination (rows stored contiguously)
//   gaddr        : global address of tile start (row-major, stride0 elements/row)
//   tensor_d0/d1 : tensor extent (elements / rows) for OOB clamping
//   tile_d0/d1   : tile extent (elements / rows)
// Descriptor packing per CDNA5 ISA 8.3/8.4 (D# group0 + group1, 2D -> groups 2/3 zero).
// ---------------------------------------------------------------------------
__device__ __forceinline__ void tdm_load_2d_f32(unsigned lds_addr, const void* gaddr,
                                                unsigned tensor_d0, unsigned tensor_d1,
                                                unsigned tile_d0, unsigned tile_d1,
                                                unsigned stride0_elems)
{
    const unsigned long long ga = (unsigned long long)(uintptr_t)gaddr;

    u32x4 g0;
    g0[0] = 1u;                                            // count=1 (valid), user mode
    g0[1] = lds_addr;                                      // lds_addr [63:32]
    g0[2] = (unsigned)(ga & 0xFFFFFFFFu);                  // global_addr[31:0]
    g0[3] = (unsigned)((ga >> 32) & 0x1FFFFFFu)            // global_addr[56:32]
          | (2u << 30);                                    // type = 2 ("image")

    i32x8 g1;
    g1[0] = (int)(2u << 16);                               // data_size=2 -> 4 bytes
    g1[1] = (int)((tensor_d0 & 0xFFFFu) << 16);            // tensor_dim0[15:0] at bits 63:48
    g1[2] = (int)(((tensor_d0 >> 16) & 0xFFFFu)            // tensor_dim0[31:16]
          |      ((tensor_d1 & 0xFFFFu) << 16));           // tensor_dim1[15:0]
    g1[3] = (int)(((tensor_d1 >> 16) & 0xFFFFu)            // tensor_dim1[31:16]
          |      ((tile_d0 & 0xFFFFu) << 16));             // tile_dim0
    g1[4] = (int)(tile_d1 & 0xFFFFu);                      // tile_dim1 (tile_dim2 = 0)
    g1[5] = (int)stride0_elems;                            // tensor_dim0_stride[31:0]
    g1[6] = 0;                                             // stride0[47:32], stride1[15:0]
    g1[7] = 0;                                             // stride1[47:16] (2D: unused)

    const i32x4 z4 = {0, 0, 0, 0};
#if __clang_major__ >= 23
    const i32x8 z8 = {0, 0, 0, 0, 0, 0, 0, 0};
    __builtin_amdgcn_tensor_load_to_lds(g0, g1, z4, z4, z8, 0);
#else
    __builtin_amdgcn_tensor_load_to_lds(g0, g1, z4, z4, 0);
#endif
}

// ---------------------------------------------------------------------------
// Kernel 2: scores[h,m] = sum_d W2[d] * relu(H[h,d] + M[m,d]) + b2 ; diag -> -inf
// Block = 256 threads computes a 32x32 tile. H/M/W2 tiles staged in LDS by the
// Tensor Data Mover (one tensor_load_to_lds per tile, s_wait_tensorcnt, barrier).
// Compute: thread owns m = lane and h = wid + {0,8,16,24}. The d-loop is skewed
// by lane ((d4 + lane) & 127) so concurrent lanes touch disjoint LDS bank groups
// (stride-512 rows, banks 4*(d4+lane) mod 64) -> conflict-free b128 reads.
// ---------------------------------------------------------------------------
__global__ __launch_bounds__(256) void pair_score(
    const float* __restrict__ H,
    const float* __restrict__ M,
    const float* __restrict__ W2,
    const float* __restrict__ b2p,
    float* __restrict__ out)
{
    __shared__ float Hs[32 * 512];
    __shared__ float Ms[32 * 512];
    __shared__ float W2s[512];

    const int tid = threadIdx.x;
    const int h0 = blockIdx.y * 32;
    const int m0 = blockIdx.x * 32;

    if (tid < 32) {  // wave 0 drives the TDM (EXEC-independent, wave-level op)
        tdm_load_2d_f32((unsigned)(uintptr_t)Hs,  H + (size_t)h0 * HIDD,
                        HIDD, 32, HIDD, 32, HIDD);
        tdm_load_2d_f32((unsigned)(uintptr_t)Ms,  M + (size_t)m0 * HIDD,
                        HIDD, 32, HIDD, 32, HIDD);
        tdm_load_2d_f32((unsigned)(uintptr_t)W2s, W2,
                        HIDD, 1, HIDD, 1, HIDD);
        __builtin_amdgcn_s_wait_tensorcnt(0);
    }
    __syncthreads();

    const float b2 = b2p[0];
    const int wid  = tid >> 5;
    const int lane = tid & 31;

    const float* ms  = &Ms[lane * 512];
    const float* hp0 = &Hs[(wid +  0) * 512];
    const float* hp1 = &Hs[(wid +  8) * 512];
    const float* hp2 = &Hs[(wid + 16) * 512];
    const float* hp3 = &Hs[(wid + 24) * 512];

    float a0 = b2, a1 = b2, a2 = b2, a3 = b2;

#pragma unroll 4
    for (int d4 = 0; d4 < 128; ++d4) {
        const int off = ((d4 + lane) & 127) << 2;   // lane-skewed d, bank-conflict-free
        const float4 wv  = *(const float4*)&W2s[off];
        const float4 mv  = *(const float4*)&ms[off];
        const float4 h0v = *(const float4*)&hp0[off];
        const float4 h1v = *(const float4*)&hp1[off];
        const float4 h2v = *(const float4*)&hp2[off];
        const float4 h3v = *(const float4*)&hp3[off];

        a0 = fmaf(wv.x, fmaxf(h0v.x + mv.x, 0.f), a0);
        a0 = fmaf(wv.y, fmaxf(h0v.y + mv.y, 0.f), a0);
        a0 = fmaf(wv.z, fmaxf(h0v.z + mv.z, 0.f), a0);
        a0 = fmaf(wv.w, fmaxf(h0v.w + mv.w, 0.f), a0);

        a1 = fmaf(wv.x, fmaxf(h1v.x + mv.x, 0.f), a1);
        a1 = fmaf(wv.y, fmaxf(h1v.y + mv.y, 0.f), a1);
        a1 = fmaf(wv.z, fmaxf(h1v.z + mv.z, 0.f), a1);
        a1 = fmaf(wv.w, fmaxf(h1v.w + mv.w, 0.f), a1);

        a2 = fmaf(wv.x, fmaxf(h2v.x + mv.x, 0.f), a2);
        a2 = fmaf(wv.y, fmaxf(h2v.y + mv.y, 0.f), a2);
        a2 = fmaf(wv.z, fmaxf(h2v.z + mv.z, 0.f), a2);
        a2 = fmaf(wv.w, fmaxf(h2v.w + mv.w, 0.f), a2);

        a3 = fmaf(wv.x, fmaxf(h3v.x + mv.x, 0.f), a3);
        a3 = fmaf(wv.y, fmaxf(h3v.y + mv.y, 0.f), a3);
        a3 = fmaf(wv.z, fmaxf(h3v.z + mv.z, 0.f), a3);
        a3 = fmaf(wv.w, fmaxf(h3v.w + mv.w, 0.f), a3);
    }

    const float ninf = -__builtin_inff();
    const int m = m0 + lane;
    const int hA = h0 + wid +  0;
    const int hB = h0 + wid +  8;
    const int hC = h0 + wid + 16;
    const int hD = h0 + wid + 24;
    out[(size_t)hA * NSAMP + m] = (hA == m) ? ninf : a0;
    out[(size_t)hB * NSAMP + m] = (hB == m) ? ninf : a1;
    out[(size_t)hC * NSAMP + m] = (hC == m) ? ninf : a2;
    out[(size_t)hD * NSAMP + m] = (hD == m) ? ninf : a3;
}

extern "C" void kernel_launch(void* const* d_in, const int* in_sizes, int n_in,
                              void* d_out, int out_size, void* d_ws, size_t ws_size,
                              hipStream_t stream) {
    const float* lstm_out = (const float*)d_in[0];
    const float* W_h      = (const float*)d_in[1];
    const float* b_h      = (const float*)d_in[2];
    const float* W_m      = (const float*)d_in[3];
    const float* W2       = (const float*)d_in[4];
    const float* b2       = (const float*)d_in[5];
    float* out = (float*)d_out;

    float* Hbuf = (float*)d_ws;                 // 512*512 f32 = 1 MB
    float* Mbuf = Hbuf + (size_t)NSAMP * HIDD;  // 512*512 f32 = 1 MB

    // Kernel 1: grid (j-strips of 64, i-tiles of 16, {H, M}), 1 wave per block
    dim3 g1(HIDD / 64, NSAMP / 16, 2);
    gemm_wmma_f32<<<g1, 32, 0, stream>>>(lstm_out, W_h, W_m, b_h, Hbuf, Mbuf);

    // Kernel 2: 32x32 output tiles
    dim3 g2(NSAMP / 32, NSAMP / 32, 1);
    pair_score<<<g2, 256, 0, stream>>>(Hbuf, Mbuf, W2, b2, out);
}